// _LSTMBase_58308476011034
// MI455X (gfx1250) — compile-verified
//
#include <hip/hip_runtime.h>
#include <hip/hip_bf16.h>

typedef float v2f __attribute__((ext_vector_type(2)));
typedef float v8f __attribute__((ext_vector_type(8)));

#define B_  32
#define T_  2048
#define I_  256
#define H_  512
#define G4  2048   // 4*H

// ---------------------------------------------------------------------------
// init: hbuf[0] <- h0, barrier counter <- 0   (runs every call; graph-safe)
// ---------------------------------------------------------------------------
__global__ void k_init(const float* __restrict__ h0,
                       float* __restrict__ hbuf,
                       unsigned* __restrict__ ctr) {
  int i = blockIdx.x * blockDim.x + threadIdx.x;
  if (i < B_ * H_) hbuf[i] = h0[i];
  if (i == 0) *ctr = 0u;
}

// ---------------------------------------------------------------------------
// Phase 1: xg[t][b][g] = sum_i x[b][t][i] * Wih[g][i] + bih[g] + bhh[g]
// Block = 128 threads (4 waves). Each block: 64 gate cols (4 N-tiles), 8 t's.
// Grid = (T/8) * (2048/64) = 256*32 = 8192 blocks.
// ---------------------------------------------------------------------------
__global__ __launch_bounds__(128) void k_xgates(
    const float* __restrict__ x, const float* __restrict__ Wih,
    const float* __restrict__ bih, const float* __restrict__ bhh,
    float* __restrict__ xg) {
  __shared__ float Ws[64 * 260];   // Wih slice, row = gate col, padded stride
  __shared__ float xs[32 * 260];   // x tile for one t, padded stride
  __shared__ float bs[64];

  const int tid = threadIdx.x;
  const int cg  = blockIdx.x & 31;   // column group
  const int tg  = blockIdx.x >> 5;   // t group
  const int g0  = cg * 64;
  const int t0  = tg * 8;

  // Stage W_ih rows g0..g0+63 (coalesced float4)
  for (int idx = tid; idx < 64 * 64; idx += 128) {
    const int r = idx >> 6, c4 = idx & 63;
    const float4 v = ((const float4*)(Wih + (size_t)(g0 + r) * I_))[c4];
    float* dst = &Ws[r * 260 + c4 * 4];
    dst[0] = v.x; dst[1] = v.y; dst[2] = v.z; dst[3] = v.w;
  }
  if (tid < 64) bs[tid] = bih[g0 + tid] + bhh[g0 + tid];

  const int lane = tid & 31;
  const int half = lane >> 4;       // 0: K=0,1   1: K=2,3
  const int l16  = lane & 15;
  const int ni   = tid >> 5;        // wave id -> N tile 0..3

  for (int tt = 0; tt < 8; ++tt) {
    const int t = t0 + tt;
    __syncthreads();                // xs reusable; Ws/bs visible on first iter
    for (int idx = tid; idx < 32 * 64; idx += 128) {
      const int r = idx >> 6, c4 = idx & 63;
      const float4 v = ((const float4*)(x + ((size_t)r * T_ + t) * I_))[c4];
      float* dst = &xs[r * 260 + c4 * 4];
      dst[0] = v.x; dst[1] = v.y; dst[2] = v.z; dst[3] = v.w;
    }
    __syncthreads();

    for (int mi = 0; mi < 2; ++mi) {
      const float bv = bs[ni * 16 + l16];          // bias depends on N only
      v8f acc = { bv, bv, bv, bv, bv, bv, bv, bv };
      const int arow = mi * 16 + l16;
      const int bcol = ni * 16 + l16;
      #pragma unroll 4
      for (int k0 = 0; k0 < I_; k0 += 4) {
        const int ka = k0 + half * 2;
        const v2f a = *(const v2f*)&xs[arow * 260 + ka];
        const v2f b = *(const v2f*)&Ws[bcol * 260 + ka];
        acc = __builtin_amdgcn_wmma_f32_16x16x4_f32(
            false, a, false, b, (short)0, acc, false, false);
      }
      const int gcol = g0 + ni * 16 + l16;
      #pragma unroll
      for (int r = 0; r < 8; ++r) {
        const int brow = mi * 16 + half * 8 + r;   // C layout: M = r + 8*half
        xg[((size_t)t * B_ + brow) * G4 + gcol] = acc[r];
      }
    }
  }
}

// ---------------------------------------------------------------------------
// Phase 2: persistent recurrence. 64 blocks x 128 threads; block w owns
// H-columns [8w, 8w+8) -> packed gate cols g(n) = (n/8)*H + 8w + n%8.
// Per step: stage h -> LDS, S = h @ Wl via v_wmma_f32_16x16x4_f32,
// gates -> c,h update, grid barrier.
// ---------------------------------------------------------------------------
__global__ __launch_bounds__(128) void k_recurrent(
    const float* __restrict__ Whh, const float* __restrict__ c0,
    const float* __restrict__ xg,  float* __restrict__ hbuf,
    unsigned* __restrict__ ctr,    float* __restrict__ out,
    float* __restrict__ out_hf,    float* __restrict__ out_cf) {
  __shared__ float Wl[32 * 516];   // packed W_hh slice, [n][k], padded stride
  __shared__ float hs[32 * 516];   // staged h, [b][k], padded stride
  __shared__ float Ssh[32 * 33];   // S tile for gate recombine

  const int tid   = threadIdx.x;
  const int w     = blockIdx.x;    // 0..63
  const int hbase = w * 8;

  // Pack W_hh slice once (coalesced float4 per gate row)
  for (int idx = tid; idx < 32 * 128; idx += 128) {
    const int n = idx >> 7, k4 = idx & 127;
    const int g = (n >> 3) * H_ + hbase + (n & 7);
    const float4 v = ((const float4*)(Whh + (size_t)g * H_))[k4];
    float* dst = &Wl[n * 516 + k4 * 4];
    dst[0] = v.x; dst[1] = v.y; dst[2] = v.z; dst[3] = v.w;
  }

  // Elementwise ownership: pairs p = tid, tid+128; b = p/8, hc = p%8
  const int eb0  = tid >> 3;        // 0..15
  const int eb1  = eb0 + 16;
  const int ehc  = tid & 7;
  const int ecol = hbase + ehc;
  float c_0 = c0[eb0 * H_ + ecol];
  float c_1 = c0[eb1 * H_ + ecol];

  const int lane = tid & 31;
  const int half = lane >> 4;
  const int l16  = lane & 15;
  const int mi   = (tid >> 5) & 1;  // wave -> output tile
  const int ni   = tid >> 6;
  const int arow = mi * 16 + l16;
  const int bcol = ni * 16 + l16;

  for (int t = 0; t < T_; ++t) {
    const float* hsrc = hbuf + (size_t)(t & 1) * (B_ * H_);
    float*       hdst = hbuf + (size_t)((t + 1) & 1) * (B_ * H_);

    // Stage h (coalesced)
    for (int idx = tid; idx < 32 * 128; idx += 128) {
      const int r = idx >> 7, k4 = idx & 127;
      const float4 v = ((const float4*)(hsrc + (size_t)r * H_))[k4];
      float* dst = &hs[r * 516 + k4 * 4];
      dst[0] = v.x; dst[1] = v.y; dst[2] = v.z; dst[3] = v.w;
    }
    __syncthreads();

    // S = h @ W_slice  (K = 512, 128 chained f32 WMMAs per wave)
    v8f acc = { 0.f, 0.f, 0.f, 0.f, 0.f, 0.f, 0.f, 0.f };
    #pragma unroll 4
    for (int k0 = 0; k0 < H_; k0 += 4) {
      const int ka = k0 + half * 2;
      const v2f a = *(const v2f*)&hs[arow * 516 + ka];
      const v2f b = *(const v2f*)&Wl[bcol * 516 + ka];
      acc = __builtin_amdgcn_wmma_f32_16x16x4_f32(
          false, a, false, b, (short)0, acc, false, false);
    }
    #pragma unroll
    for (int r = 0; r < 8; ++r)
      Ssh[(mi * 16 + half * 8 + r) * 33 + ni * 16 + l16] = acc[r];
    __syncthreads();

    // Gate math for this thread's two (b, hcol) pairs
    {
      const float* xgt = xg + (size_t)t * B_ * G4;
      float hv[2]; float* cc[2] = { &c_0, &c_1 }; const int bb[2] = { eb0, eb1 };
      #pragma unroll
      for (int p = 0; p < 2; ++p) {
        const int b = bb[p];
        const float si = Ssh[b * 33 +      ehc] + xgt[(size_t)b * G4 +           ecol];
        const float sf = Ssh[b * 33 +  8 + ehc] + xgt[(size_t)b * G4 +     H_ + ecol];
        const float sg = Ssh[b * 33 + 16 + ehc] + xgt[(size_t)b * G4 + 2 * H_ + ecol];
        const float so = Ssh[b * 33 + 24 + ehc] + xgt[(size_t)b * G4 + 3 * H_ + ecol];
        const float ig = 1.f / (1.f + __expf(-si));
        const float fg = 1.f / (1.f + __expf(-sf));
        const float gg = tanhf(sg);
        const float og = 1.f / (1.f + __expf(-so));
        const float cn = fg * (*cc[p]) + ig * gg;
        *cc[p] = cn;
        hv[p] = og * tanhf(cn);
        hdst[b * H_ + ecol] = hv[p];
        out[((size_t)b * T_ + t) * H_ + ecol] = hv[p];
      }
      if (t == T_ - 1) {
        out_hf[eb0 * H_ + ecol] = hv[0];
        out_hf[eb1 * H_ + ecol] = hv[1];
        out_cf[eb0 * H_ + ecol] = c_0;
        out_cf[eb1 * H_ + ecol] = c_1;
      }
    }

    // Grid-wide barrier (monotonic counter; reset each call by k_init)
    __threadfence();
    __syncthreads();
    if (tid == 0) {
      atomicAdd(ctr, 1u);
      const unsigned target = (unsigned)gridDim.x * (unsigned)(t + 1);
      while (__hip_atomic_load(ctr, __ATOMIC_RELAXED,
                               __HIP_MEMORY_SCOPE_AGENT) < target)
        __builtin_amdgcn_s_sleep(2);
    }
    __syncthreads();
    __threadfence();
  }
}

// ---------------------------------------------------------------------------
extern "C" void kernel_launch(void* const* d_in, const int* in_sizes, int n_in,
                              void* d_out, int out_size, void* d_ws, size_t ws_size,
                              hipStream_t stream) {
  const float* x    = (const float*)d_in[0];
  const float* h0   = (const float*)d_in[1];
  const float* c0   = (const float*)d_in[2];
  const float* Wih  = (const float*)d_in[3];
  const float* Whh  = (const float*)d_in[4];
  const float* bih  = (const float*)d_in[5];
  const float* bhh  = (const float*)d_in[6];

  float* out    = (float*)d_out;                       // [B,T,H]
  float* out_hf = out + (size_t)B_ * T_ * H_;          // [1,B,H]
  float* out_cf = out_hf + (size_t)B_ * H_;            // [1,B,H]

  // Workspace: xg [T,B,4H] f32 (512 MB) + h double buffer + barrier counter
  float*    xg   = (float*)d_ws;
  float*    hbuf = xg + (size_t)T_ * B_ * G4;
  unsigned* ctr  = (unsigned*)(hbuf + 2 * (size_t)B_ * H_);

  k_init<<<(B_ * H_ + 255) / 256, 256, 0, stream>>>(h0, hbuf, ctr);

  k_xgates<<<(T_ / 8) * (G4 / 64), 128, 0, stream>>>(x, Wih, bih, bhh, xg);

  k_recurrent<<<H_ / 8, 128, 0, stream>>>(Whh, c0, xg, hbuf, ctr,
                                          out, out_hf, out_cf);
}